// SGC_90486370992433
// MI455X (gfx1250) — compile-verified
//
#include <hip/hip_runtime.h>

#define NNODES 100000
#define NF 64
#define NC 64

typedef float v2f __attribute__((ext_vector_type(2)));
typedef float v8f __attribute__((ext_vector_type(8)));

// ---------------------------------------------------------------------------
// Zero a float buffer (float4-wide).
// ---------------------------------------------------------------------------
__global__ void SGC_zero_f4(float4* __restrict__ p, int n4) {
    int i = blockIdx.x * blockDim.x + threadIdx.x;
    if (i < n4) p[i] = make_float4(0.f, 0.f, 0.f, 0.f);
}

// ---------------------------------------------------------------------------
// Edge-parallel SpMM: hout[row[e], :] += w[e] * hin[col[e], :]
// 16 lanes per edge, 4 contiguous feats per lane (float4 gather, L2-resident).
// ---------------------------------------------------------------------------
__global__ void SGC_spmm(const int* __restrict__ erow, const int* __restrict__ ecol,
                         const float* __restrict__ ew, const float* __restrict__ hin,
                         float* __restrict__ hout, int E) {
    int t = blockIdx.x * blockDim.x + threadIdx.x;
    int e = t >> 4;
    if (e >= E) return;
    int f = (t & 15) << 2;
    int r = erow[e];
    int c = ecol[e];
    float wt = ew[e];
    const float4 v = *reinterpret_cast<const float4*>(hin + (long)c * NF + f);
    float* dst = hout + (long)r * NF + f;
    atomicAdd(dst + 0, wt * v.x);
    atomicAdd(dst + 1, wt * v.y);
    atomicAdd(dst + 2, wt * v.z);
    atomicAdd(dst + 3, wt * v.w);
}

// ---------------------------------------------------------------------------
// In-place linear layer via V_WMMA_F32_16X16X4_F32:
//   h[16x64 tile] <- h[16x64] @ W^T + b
// One wave owns 16 full rows -> reads complete before its stores, and no other
// wave touches those rows, so aliasing h == out is safe.
//
// Fragment layouts (CDNA5 ISA 7.12.2):
//   A 16x4 f32 : lane L holds row M=L%16; VGPR0/1 = K={0,1} (L<16) or {2,3}.
//   B 4x16 f32 : lane L holds col N=L%16; VGPR0/1 = K={0,1} (L<16) or {2,3}.
//   C/D 16x16  : VGPR r, lanes 0-15 -> M=r, lanes 16-31 -> M=r+8, N=L%16.
// ---------------------------------------------------------------------------
__global__ void SGC_linear_wmma(float* __restrict__ h,
                                const float* __restrict__ W,
                                const float* __restrict__ bias) {
    const int lane  = threadIdx.x & 31;
    const int wave  = (blockIdx.x * blockDim.x + threadIdx.x) >> 5;
    const int row0  = wave * 16;
    if (row0 >= NNODES) return;          // uniform per wave: EXEC stays all-1s

    const int m     = lane & 15;         // A row / B col / D col within tile
    const int khalf = (lane >> 4) * 2;   // K pair selector: {0,1} vs {2,3}
    const int chalf = (lane >> 4) * 8;   // D row offset for upper half-wave

    // Seed accumulators with the bias (out[:, n] += b[n]).
    v8f acc[4];
#pragma unroll
    for (int nt = 0; nt < 4; ++nt) {
        float bv = bias[nt * 16 + m];
#pragma unroll
        for (int r = 0; r < 8; ++r) acc[nt][r] = bv;
    }

    const float* arow = h + (long)(row0 + m) * NF + khalf;
#pragma unroll
    for (int k0 = 0; k0 < NF; k0 += 4) {
        v2f a;
        a.x = arow[k0];
        a.y = arow[k0 + 1];
#pragma unroll
        for (int nt = 0; nt < 4; ++nt) {
            // out[i][n] = sum_k h[i][k] * W[n][k]  -> B(k,n) = W[col0+n][k]
            const float* wrow = W + (long)(nt * 16 + m) * NF + k0 + khalf;
            v2f bm;
            bm.x = wrow[0];
            bm.y = wrow[1];
            acc[nt] = __builtin_amdgcn_wmma_f32_16x16x4_f32(
                /*neg_a=*/false, a, /*neg_b=*/false, bm,
                /*c_mod=*/(short)0, acc[nt], /*reuse_a=*/false, /*reuse_b=*/false);
        }
    }

    // Store D tiles (in-place).
#pragma unroll
    for (int nt = 0; nt < 4; ++nt)
#pragma unroll
        for (int r = 0; r < 8; ++r)
            h[(long)(row0 + chalf + r) * NF + nt * 16 + m] = acc[nt][r];
}

// ---------------------------------------------------------------------------
// Launch: zero(ws) -> spmm(x->ws) -> zero(out) -> spmm(ws->out) -> linear(out)
// k is the static propagation depth (=2 in the reference).
// ---------------------------------------------------------------------------
extern "C" void kernel_launch(void* const* d_in, const int* in_sizes, int n_in,
                              void* d_out, int out_size, void* d_ws, size_t ws_size,
                              hipStream_t stream) {
    const float* x    = (const float*)d_in[0];
    const int*   erow = (const int*)d_in[1];
    const int*   ecol = (const int*)d_in[2];
    const float* ew   = (const float*)d_in[3];
    const float* Wm   = (const float*)d_in[4];
    const float* bv   = (const float*)d_in[5];
    const int    E    = in_sizes[1];

    float* h1 = (float*)d_ws;    // intermediate A@x   (25.6 MB)
    float* h2 = (float*)d_out;   // A@(A@x), then overwritten in-place by linear

    const int n4 = NNODES * NF / 4;
    dim3 blk(256);

    SGC_zero_f4<<<(n4 + 255) / 256, blk, 0, stream>>>((float4*)h1, n4);
    {
        long t = (long)E * 16;
        SGC_spmm<<<(unsigned)((t + 255) / 256), blk, 0, stream>>>(erow, ecol, ew, x, h1, E);
    }
    SGC_zero_f4<<<(n4 + 255) / 256, blk, 0, stream>>>((float4*)h2, n4);
    {
        long t = (long)E * 16;
        SGC_spmm<<<(unsigned)((t + 255) / 256), blk, 0, stream>>>(erow, ecol, ew, h1, h2, E);
    }
    {
        const int waves = NNODES / 16;              // 6250 tiles
        const int threads = waves * 32;
        SGC_linear_wmma<<<(threads + 255) / 256, blk, 0, stream>>>(h2, Wm, bv);
    }
}